// IDLoss_84670985274125
// MI455X (gfx1250) — compile-verified
//
#include <hip/hip_runtime.h>
#include <math.h>

// Problem constants (fixed by the reference).
#define G      256
#define NPAIRS 32896.0f   // (257*256)/2 pairs over G+1 groups (phantom group appended)

typedef __attribute__((ext_vector_type(2))) float v2f;
typedef __attribute__((ext_vector_type(8))) float v8f;

// ---------------------------------------------------------------------------
// Kernel 0: zero the 768-float accumulator region of the workspace.
// ---------------------------------------------------------------------------
__global__ void idloss_zero_ws(float* __restrict__ ws) {
    int t = threadIdx.x;           // 256 threads
    ws[t]       = 0.0f;            // cnt
    ws[t + 256] = 0.0f;            // sum(x)
    ws[t + 512] = 0.0f;            // sum(x^2)
}

// ---------------------------------------------------------------------------
// Kernel 1: one-pass fused segment reduction (cnt, s1, s2) into ws.
// Memory-bound: streams 128 MB once via float4/int4 loads; per-block LDS
// histograms via ds_add_f32; single global atomic flush per block.
// ---------------------------------------------------------------------------
__global__ __launch_bounds__(256) void idloss_seg_stats(
        const float* __restrict__ pred,
        const int*   __restrict__ tgt,
        float*       __restrict__ ws,
        int n) {
    __shared__ float scnt[G];
    __shared__ float ss1[G];
    __shared__ float ss2[G];

    const int t = threadIdx.x;
    scnt[t] = 0.0f; ss1[t] = 0.0f; ss2[t] = 0.0f;
    __syncthreads();

    const int n4     = n >> 2;
    const int stride = gridDim.x * blockDim.x;
    const float4* __restrict__ p4 = (const float4*)pred;
    const int4*   __restrict__ g4 = (const int4*)tgt;

    for (int i = blockIdx.x * blockDim.x + t; i < n4; i += stride) {
        // Speculative prefetch of the next grid-stride chunk (global_prefetch_b8).
        __builtin_prefetch(&p4[i + stride], 0, 0);
        __builtin_prefetch(&g4[i + stride], 0, 0);

        float4 p = p4[i];
        int4   g = g4[i];

        atomicAdd(&ss1[g.x], p.x); atomicAdd(&ss2[g.x], p.x * p.x); atomicAdd(&scnt[g.x], 1.0f);
        atomicAdd(&ss1[g.y], p.y); atomicAdd(&ss2[g.y], p.y * p.y); atomicAdd(&scnt[g.y], 1.0f);
        atomicAdd(&ss1[g.z], p.z); atomicAdd(&ss2[g.z], p.z * p.z); atomicAdd(&scnt[g.z], 1.0f);
        atomicAdd(&ss1[g.w], p.w); atomicAdd(&ss2[g.w], p.w * p.w); atomicAdd(&scnt[g.w], 1.0f);
    }

    // Scalar tail (n not multiple of 4) handled by the first few global threads.
    const int gid  = blockIdx.x * blockDim.x + t;
    const int tail = n - (n4 << 2);
    if (gid < tail) {
        int   idx = (n4 << 2) + gid;
        float p   = pred[idx];
        int   g   = tgt[idx];
        atomicAdd(&ss1[g], p); atomicAdd(&ss2[g], p * p); atomicAdd(&scnt[g], 1.0f);
    }

    __syncthreads();
    atomicAdd(&ws[t],       scnt[t]);
    atomicAdd(&ws[t + 256], ss1[t]);
    atomicAdd(&ws[t + 512], ss2[t]);
}

// ---------------------------------------------------------------------------
// Kernel 2: per-bin mean/std, pairwise v_iou sum over 257 groups, final
// reduction via V_WMMA_F32_16X16X4_F32 ones-matmul + wave shfl reduction.
// ---------------------------------------------------------------------------
__global__ __launch_bounds__(256) void idloss_pair_loss(
        const float* __restrict__ ws,
        float*       __restrict__ out) {
    __shared__ float mean[G + 1];
    __shared__ float sd[G + 1];
    __shared__ float part[256];

    const int t = threadIdx.x;

    // Per-bin statistics. ss = s2 - s1^2/cnt == sum((x-mean)^2) exactly.
    float c  = ws[t];
    float s1 = ws[t + 256];
    float s2 = ws[t + 512];
    float m  = s1 / c;
    float var = (s2 - s1 * s1 / c) / fmaxf(c - 1.0f, 1.0f);
    float st  = (c > 1.0f) ? sqrtf(fmaxf(var, 0.0f)) : 0.0f;
    mean[t] = m;
    sd[t]   = st;
    if (t == 0) { mean[G] = 0.0f; sd[G] = 0.0f; }  // phantom group
    __syncthreads();

    // Row i = t, columns j = i+1 .. G. Total terms = 32,896 (tiny).
    float acc = 0.0f;
    const float mi = mean[t];
    const float si = sd[t];
    for (int j = t + 1; j <= G; ++j) {
        float d = fabsf(mean[j] - mi);
        float s = si + sd[j];
        acc += s / (d + s);
    }
    part[t] = acc;
    __syncthreads();

    // Wave 0: reduce the 256 partials with 4 chained V_WMMA_F32_16X16X4_F32.
    // A = ones(16x4), B[k,n] = part[chunk*64 + k*16 + n]  =>  D[m,n] += col-sums.
    // A/B fragments (ISA 7.12.2, 32-bit 16x4 / 4x16): lane<16 holds K={0,1},
    // lane>=16 holds K={2,3}; column/row index = lane & 15.
    if (t < 32) {
        const int lane = t;
        const int half = lane >> 4;   // 0: K=0,1   1: K=2,3
        const int l    = lane & 15;

        v2f a; a[0] = 1.0f; a[1] = 1.0f;
        v8f cacc = {};
        for (int chunk = 0; chunk < 4; ++chunk) {
            int base = chunk * 64 + half * 32 + l;
            v2f b;
            b[0] = part[base];        // K = 2*half
            b[1] = part[base + 16];   // K = 2*half + 1
            cacc = __builtin_amdgcn_wmma_f32_16x16x4_f32(
                false, a, false, b, (short)0, cacc,
                /*reuse_a=*/false, /*reuse_b=*/false);
        }

        // D[m,n] is identical for all m (A all ones); each lane's cacc[0] is
        // the column sum for n = lane&15, duplicated across the two halves.
        // Full-wave xor reduction therefore yields 2x the total.
        float v = cacc[0];
        for (int off = 16; off > 0; off >>= 1)
            v += __shfl_xor(v, off, 32);

        if (lane == 0)
            out[0] = (0.5f * v) / NPAIRS;
    }
}

// ---------------------------------------------------------------------------
extern "C" void kernel_launch(void* const* d_in, const int* in_sizes, int n_in,
                              void* d_out, int out_size, void* d_ws, size_t ws_size,
                              hipStream_t stream) {
    const float* pred = (const float*)d_in[0];
    const int*   tgt  = (const int*)d_in[1];
    float*       ws   = (float*)d_ws;
    float*       out  = (float*)d_out;
    const int    n    = in_sizes[0];

    idloss_zero_ws<<<1, 256, 0, stream>>>(ws);

    // 2048 blocks x 256 threads (8 wave32/WGP-group): ~8 float4 iters/thread.
    idloss_seg_stats<<<2048, 256, 0, stream>>>(pred, tgt, ws, n);

    idloss_pair_loss<<<1, 256, 0, stream>>>(ws, out);
}